// LearnableLaplacianBasis_12670153523565
// MI455X (gfx1250) — compile-verified
//
#include <hip/hip_runtime.h>
#include <math.h>

// ---------------- problem constants (match reference) ----------------
#define K_DIM   64
#define D_DIM   256
#define B_DIM   16
#define T_DIM   8192
#define T_CHUNK 128
#define N_CHUNKS (T_DIM / T_CHUNK)   // 64
#define XS_STRIDE 260                // 256+4 pad: conflict-free A-frag b64 column reads
#define WP_STRIDE 160                // 128+32 pad: stride ≡ 32 (mod 64 banks) -> the two
                                     // half-waves (rows dp, dp+1) hit disjoint bank halves
#define Z_STRIDE  132                // 128 + 4 pad

// ---------------- workspace layout (in floats) ----------------
#define WS_WPK    0                                   // packed W: [128 dpair][64 n][2] = 16384
#define WS_BIAS   (WS_WPK + D_DIM*K_DIM)              // 64
#define WS_LA     (WS_BIAS + K_DIM)                   // 64  (alpha0*dt)
#define WS_TH     (WS_LA + K_DIM)                     // 64  (theta)
#define WS_ARE    (WS_TH + K_DIM)                     // 64  Re(a)
#define WS_AIM    (WS_ARE + K_DIM)                    // 64  Im(a)
#define WS_E      (WS_AIM + K_DIM)                    // chunk-end states: B*NC*K*2
#define WS_CARRY  (WS_E + B_DIM*N_CHUNKS*K_DIM*2)     // chunk carries:    B*NC*K*2

typedef float v2f __attribute__((ext_vector_type(2)));
typedef float v8f __attribute__((ext_vector_type(8)));
typedef int   v4i_gcc __attribute__((vector_size(16)));   // matches builtin's "V4i"

// ---- CDNA5 async global->LDS path (guarded: falls back to plain loads) ----
#if __has_builtin(__builtin_amdgcn_global_load_async_to_lds_b128) && \
    __has_builtin(__builtin_amdgcn_s_wait_asynccnt)
#define HAVE_ASYNC_LDS 1
#else
#define HAVE_ASYNC_LDS 0
#endif

#if HAVE_ASYNC_LDS
typedef __attribute__((address_space(1))) v4i_gcc* gv4i_p;   // global int4*
typedef __attribute__((address_space(3))) v4i_gcc* lv4i_p;   // LDS int4*
__device__ static inline void async_copy16(const float* gsrc, float* ldst) {
  // per-lane: 16 bytes global -> LDS, tracked by ASYNCcnt (no VGPR data path)
  __builtin_amdgcn_global_load_async_to_lds_b128(
      (gv4i_p)(uintptr_t)gsrc,
      (lv4i_p)ldst,
      0, 0);
}
#endif

// ============================================================================
// Kernel 1: spectral-norm power iteration + scalar precompute (single block)
// ============================================================================
__global__ void __launch_bounds__(256) k_setup(const float* __restrict__ s_real,
                                               const float* __restrict__ s_imag,
                                               const float* __restrict__ tau_raw,
                                               const float* __restrict__ W,
                                               const float* __restrict__ bvec,
                                               const float* __restrict__ u0,
                                               float* __restrict__ ws) {
  __shared__ float u_s[K_DIM];
  __shared__ float v_s[D_DIM];
  __shared__ float red[256];
  const int tid = threadIdx.x;
  const float eps = 1e-7f;

  // ---- normalize u0 ----
  float uv = (tid < K_DIM) ? u0[tid] : 0.0f;
  red[tid] = uv * uv;
  __syncthreads();
  for (int s = 128; s > 0; s >>= 1) { if (tid < s) red[tid] += red[tid + s]; __syncthreads(); }
  float nrm = sqrtf(red[0]) + eps;
  if (tid < K_DIM) u_s[tid] = uv / nrm;
  __syncthreads();

  // ---- 10 power iterations ----
  for (int it = 0; it < 10; ++it) {
    float vd = 0.0f;                                     // v = normalize(W^T u)
    for (int k = 0; k < K_DIM; ++k) vd += W[k * D_DIM + tid] * u_s[k];
    red[tid] = vd * vd;
    __syncthreads();
    for (int s = 128; s > 0; s >>= 1) { if (tid < s) red[tid] += red[tid + s]; __syncthreads(); }
    nrm = sqrtf(red[0]) + eps;
    v_s[tid] = vd / nrm;
    __syncthreads();
    float uk = 0.0f;                                     // u = normalize(W v)
    if (tid < K_DIM) {
      const float* wr = W + tid * D_DIM;
      for (int d = 0; d < D_DIM; ++d) uk += wr[d] * v_s[d];
    }
    red[tid] = (tid < K_DIM) ? uk * uk : 0.0f;
    __syncthreads();
    for (int s = 128; s > 0; s >>= 1) { if (tid < s) red[tid] += red[tid + s]; __syncthreads(); }
    nrm = sqrtf(red[0]) + eps;
    if (tid < K_DIM) u_s[tid] = uk / nrm;
    __syncthreads();
  }

  // ---- sigma = u . (W v),  v = normalize(W^T u) ----
  float vd = 0.0f;
  for (int k = 0; k < K_DIM; ++k) vd += W[k * D_DIM + tid] * u_s[k];
  red[tid] = vd * vd;
  __syncthreads();
  for (int s = 128; s > 0; s >>= 1) { if (tid < s) red[tid] += red[tid + s]; __syncthreads(); }
  nrm = sqrtf(red[0]) + eps;
  v_s[tid] = vd / nrm;
  __syncthreads();
  float uk = 0.0f;
  if (tid < K_DIM) {
    const float* wr = W + tid * D_DIM;
    for (int d = 0; d < D_DIM; ++d) uk += wr[d] * v_s[d];
  }
  red[tid] = (tid < K_DIM) ? u_s[tid] * uk : 0.0f;
  __syncthreads();
  for (int s = 128; s > 0; s >>= 1) { if (tid < s) red[tid] += red[tid + s]; __syncthreads(); }
  const float inv_sigma = 1.0f / red[0];

  // ---- per-k recurrence params ----
  if (tid < K_DIM) {
    const float tau = log1pf(expf(tau_raw[0])) + 0.001f;       // softplus + 0.001
    const float dt  = 1.0f / (float)(T_DIM - 1);
    const float la  = fmaxf(s_real[tid], 1e-6f) * tau * dt;    // alpha0*dt
    const float th  = s_imag[tid] * tau * dt;                  // theta
    const float dec = expf(-la);
    ws[WS_LA  + tid] = la;
    ws[WS_TH  + tid] = th;
    ws[WS_ARE + tid] = dec * cosf(th);
    ws[WS_AIM + tid] = dec * sinf(th);
    ws[WS_BIAS + tid] = bvec[tid];
  }
  // ---- pair-packed W: wspk[dp][n][j] = W[n][2dp+j] / sigma  (B-frag = one b64) ----
  for (int i = 0; i < (D_DIM * K_DIM) / 256; ++i) {
    const int idx = i * 256 + tid;        // 0..16383
    const int dp  = idx >> 7;             // d-pair 0..127
    const int r   = idx & 127;
    const int n   = r >> 1;
    const int jj  = r & 1;
    ws[WS_WPK + idx] = W[n * D_DIM + 2 * dp + jj] * inv_sigma;
  }
}

// ============================================================================
// Kernel 2: WMMA GEMM (u = x*Wt + b) + in-chunk complex scan (zero carry)
// Grid: (N_CHUNKS, B).  256 threads = 8 wave32, one 16-row tile per wave.
// ============================================================================
__global__ void __launch_bounds__(256) k_gemm_scan(const float* __restrict__ x,
                                                   const float* __restrict__ ws,
                                                   float* __restrict__ out,
                                                   float* __restrict__ wsE) {
  extern __shared__ float smem[];
  float* xs  = smem;                                  // [T_CHUNK][XS_STRIDE]
  float* wpk = xs + T_CHUNK * XS_STRIDE;              // [128 dpair][WP_STRIDE]
  float* zb  = wpk + 128 * WP_STRIDE;                 // [T_CHUNK][Z_STRIDE]

  const int tid   = threadIdx.x;
  const int chunk = blockIdx.x;
  const int batch = blockIdx.y;
  const int tbase = chunk * T_CHUNK;
  const float* xrow0 = x + ((size_t)batch * T_DIM + tbase) * D_DIM;
  const float* wsrc  = ws + WS_WPK;

#if HAVE_ASYNC_LDS
  // ---- async stage W (4096 x 16B) and x chunk (8192 x 16B) into LDS ----
  for (int i = 0; i < (D_DIM * K_DIM) / (256 * 4); ++i) {     // 16 iters
    const int f  = i * 256 + tid;       // float4 index
    const int dp = f >> 5;              // 32 float4 per 128-float row
    const int j  = (f & 31) << 2;
    async_copy16(wsrc + dp * 128 + j, wpk + dp * WP_STRIDE + j);
  }
  for (int i = 0; i < (T_CHUNK * D_DIM) / (256 * 4); ++i) {   // 32 iters
    const int f  = i * 256 + tid;
    const int t  = f >> 6;              // 64 float4 per 256-float row
    const int dd = (f & 63) << 2;
    async_copy16(xrow0 + (size_t)t * D_DIM + dd, xs + t * XS_STRIDE + dd);
  }
  __builtin_amdgcn_s_wait_asynccnt(0);
#else
  for (int i = 0; i < (D_DIM * K_DIM) / 256; ++i) {           // 64 iters
    const int idx = i * 256 + tid;
    const int dp  = idx >> 7;
    const int r   = idx & 127;
    wpk[dp * WP_STRIDE + r] = wsrc[idx];
  }
  for (int i = 0; i < (T_CHUNK * D_DIM) / (256 * 4); ++i) {
    const int f  = i * 256 + tid;
    const int t  = f >> 6;
    const int dd = (f & 63) << 2;
    const float4 g = *reinterpret_cast<const float4*>(xrow0 + (size_t)t * D_DIM + dd);
    float* dst = xs + t * XS_STRIDE + dd;
    dst[0] = g.x; dst[1] = g.y; dst[2] = g.z; dst[3] = g.w;
  }
#endif
  __syncthreads();

  // ---- WMMA: each wave computes its 16(t) x 64(k) tile, K-reduction over D ----
  const int lane = tid & 31;
  const int wave = tid >> 5;
  const int half = lane >> 4;   // ISA f32 frag: lanes 0-15 K={0,1}, 16-31 K={2,3}
  const int m    = lane & 15;
  const int t0   = wave * 16;

  v8f acc0 = {}, acc1 = {}, acc2 = {}, acc3 = {};
  const float* arow  = xs + (t0 + m) * XS_STRIDE + 2 * half;          // A pair base
  const float* wcol  = wpk + half * WP_STRIDE + m * 2;                // B pair base
  for (int d0 = 0; d0 < D_DIM; d0 += 4) {
    const v2f a = *reinterpret_cast<const v2f*>(arow + d0);           // one ds_load_b64
    const float* wb = wcol + (d0 >> 1) * WP_STRIDE;
    const v2f b0 = *reinterpret_cast<const v2f*>(wb);                 // n0 = 0
    const v2f b1 = *reinterpret_cast<const v2f*>(wb + 32);            // n0 = 16
    const v2f b2 = *reinterpret_cast<const v2f*>(wb + 64);            // n0 = 32
    const v2f b3 = *reinterpret_cast<const v2f*>(wb + 96);            // n0 = 48
    acc0 = __builtin_amdgcn_wmma_f32_16x16x4_f32(false, a, false, b0, (short)0, acc0, false, false);
    acc1 = __builtin_amdgcn_wmma_f32_16x16x4_f32(false, a, false, b1, (short)0, acc1, false, false);
    acc2 = __builtin_amdgcn_wmma_f32_16x16x4_f32(false, a, false, b2, (short)0, acc2, false, false);
    acc3 = __builtin_amdgcn_wmma_f32_16x16x4_f32(false, a, false, b3, (short)0, acc3, false, false);
  }

  // ---- add bias, spill u tile to LDS (C/D layout: vgpr r -> row t0+r+8*half, col n0+m)
  {
    const float b0 = ws[WS_BIAS + m];
    const float b1 = ws[WS_BIAS + 16 + m];
    const float b2 = ws[WS_BIAS + 32 + m];
    const float b3 = ws[WS_BIAS + 48 + m];
    float* zrow = zb + (t0 + 8 * half) * Z_STRIDE + m;
    for (int r = 0; r < 8; ++r) {
      float* p = zrow + r * Z_STRIDE;
      p[0]  = acc0[r] + b0;
      p[16] = acc1[r] + b1;
      p[32] = acc2[r] + b2;
      p[48] = acc3[r] + b3;
    }
  }
  __syncthreads();

  // ---- local complex scan, zero initial state (threads 0..63 = one k each) ----
  if (tid < K_DIM) {
    const int k = tid;
    const float ar = ws[WS_ARE + k];
    const float ai = ws[WS_AIM + k];
    float zr = 0.0f, zi = 0.0f;
    float* col = zb + k;
    for (int t = 0; t < T_CHUNK; ++t) {
      const float u = col[t * Z_STRIDE];
      const float nzr = fmaf(ar, zr, fmaf(-ai, zi, u));
      const float nzi = fmaf(ar, zi, ai * zr);
      zr = nzr; zi = nzi;
      col[t * Z_STRIDE]         = zr;   // real
      col[t * Z_STRIDE + K_DIM] = zi;   // imag
    }
    float2 e; e.x = zr; e.y = zi;       // chunk-end state
    *reinterpret_cast<float2*>(wsE + (((size_t)batch * N_CHUNKS + chunk) * K_DIM + k) * 2) = e;
  }
  __syncthreads();

  // ---- write local z to d_out ([B,T,2K], re|im) ----
  float* orow = out + ((size_t)batch * T_DIM + tbase) * (2 * K_DIM);
  for (int i = 0; i < (T_CHUNK * 2 * K_DIM) / 256; ++i) {
    const int f = i * 256 + tid;
    const int t = f >> 7;
    const int c = f & 127;
    orow[f] = zb[t * Z_STRIDE + c];
  }
}

// ============================================================================
// Kernel 3: carries across chunks: G_{-1}=0; carry[c]=G_{c-1}; G_c=E_c+a^L G_{c-1}
// ============================================================================
__global__ void __launch_bounds__(256) k_carry(const float* __restrict__ ws,
                                               const float* __restrict__ wsE,
                                               float* __restrict__ wsC) {
  const int g = blockIdx.x * blockDim.x + threadIdx.x;   // 0..1023
  const int b = g >> 6;
  const int k = g & 63;
  const float la = ws[WS_LA + k];
  const float th = ws[WS_TH + k];
  const float dec = expf(-la * (float)T_CHUNK);
  const float ang = th * (float)T_CHUNK;
  const float aLr = dec * cosf(ang);
  const float aLi = dec * sinf(ang);
  float Gr = 0.0f, Gi = 0.0f;
  for (int c = 0; c < N_CHUNKS; ++c) {
    const size_t idx = (((size_t)b * N_CHUNKS + c) * K_DIM + k) * 2;
    wsC[idx]     = Gr;
    wsC[idx + 1] = Gi;
    const float Er = wsE[idx], Ei = wsE[idx + 1];
    const float nGr = fmaf(aLr, Gr, fmaf(-aLi, Gi, Er));
    const float nGi = fmaf(aLr, Gi, fmaf(aLi, Gr, Ei));
    Gr = nGr; Gi = nGi;
  }
}

// ============================================================================
// Kernel 4: fixup out += a^{t+1} * carry  (grid.x = N_CHUNKS-1, chunk 0 skipped)
// ============================================================================
__global__ void __launch_bounds__(256) k_fixup(const float* __restrict__ ws,
                                               const float* __restrict__ wsC,
                                               float* __restrict__ out) {
  __shared__ float Cr[K_DIM], Ci[K_DIM], sla[K_DIM], sth[K_DIM];
  const int tid   = threadIdx.x;
  const int chunk = blockIdx.x + 1;
  const int batch = blockIdx.y;
  if (tid < K_DIM) {
    const size_t idx = (((size_t)batch * N_CHUNKS + chunk) * K_DIM + tid) * 2;
    Cr[tid] = wsC[idx];
    Ci[tid] = wsC[idx + 1];
    sla[tid] = ws[WS_LA + tid];
    sth[tid] = ws[WS_TH + tid];
  }
  __syncthreads();
  float* orow = out + ((size_t)batch * T_DIM + chunk * T_CHUNK) * (2 * K_DIM);
  for (int i = 0; i < (T_CHUNK * 2 * K_DIM) / 256; ++i) {
    const int f = i * 256 + tid;
    const int t = f >> 7;
    const int c = f & 127;
    const int k = c & 63;
    const float p  = (float)(t + 1);
    const float w  = expf(-sla[k] * p);
    const float cr = w * cosf(sth[k] * p);
    const float ci = w * sinf(sth[k] * p);
    const float fr = cr * Cr[k] - ci * Ci[k];
    const float fi = cr * Ci[k] + ci * Cr[k];
    orow[f] += (c < K_DIM) ? fr : fi;
  }
}

// ============================================================================
extern "C" void kernel_launch(void* const* d_in, const int* in_sizes, int n_in,
                              void* d_out, int out_size, void* d_ws, size_t ws_size,
                              hipStream_t stream) {
  (void)in_sizes; (void)n_in; (void)out_size; (void)ws_size;
  const float* x      = (const float*)d_in[0];
  const float* s_real = (const float*)d_in[1];
  const float* s_imag = (const float*)d_in[2];
  const float* tau    = (const float*)d_in[3];
  const float* W      = (const float*)d_in[4];
  const float* bvec   = (const float*)d_in[5];
  const float* u0     = (const float*)d_in[6];
  float* out = (float*)d_out;
  float* ws  = (float*)d_ws;   // uses ~1.1 MB of scratch

  k_setup<<<1, 256, 0, stream>>>(s_real, s_imag, tau, W, bvec, u0, ws);

  const size_t smem = (size_t)(T_CHUNK * XS_STRIDE + 128 * WP_STRIDE + T_CHUNK * Z_STRIDE)
                      * sizeof(float);                 // ~276 KB (< 320 KB/WG on CDNA5)
  k_gemm_scan<<<dim3(N_CHUNKS, B_DIM), 256, smem, stream>>>(x, ws, out, ws + WS_E);

  k_carry<<<4, 256, 0, stream>>>(ws, ws + WS_E, ws + WS_CARRY);

  k_fixup<<<dim3(N_CHUNKS - 1, B_DIM), 256, 0, stream>>>(ws, ws + WS_CARRY, out);
}